// DLRM_Net_70781061038446
// MI455X (gfx1250) — compile-verified
//
#include <hip/hip_runtime.h>
#include <hip/hip_bf16.h>
#include <math.h>

// ---------------------------------------------------------------------------
// DLRM batch-1 forward on gfx1250 (CDNA5, wave32).
//   d_in[0]  dense_x   [1,1,13]   f32
//   d_in[1]  lS_i      [26,100]   int (harness maps integer -> const int*)
//   d_in[2]  emb_w     [26,100000,128] f32
//   d_in[3..8]   bottom MLP (W,b) x3 : (512,13),(512) (256,512),(256) (128,256),(128)
//   d_in[9..16]  top MLP (W,b) x4    : (1024,479),(1024,1024),(512,1024),(1,512)
//   d_out    [1,1] f32
// ---------------------------------------------------------------------------

typedef float v2f_t __attribute__((ext_vector_type(2)));
typedef float v8f_t __attribute__((ext_vector_type(8)));
typedef unsigned int u32x4_t __attribute__((ext_vector_type(4)));
typedef int          i32x4_t __attribute__((ext_vector_type(4)));
typedef int          i32x8_t __attribute__((ext_vector_type(8)));

#define NTAB   26
#define NROWS  100000
#define EMB_D  128
#define POOL_L 100

#define CHUNK_K        64     // K columns per TDM tile
#define ROWS_PER_BLK   64     // W rows (outputs) per block
#define LDS_STRIDE     66     // 64 data dwords + 2 pad dwords (bank-conflict-free)

// ---------------------------------------------------------------------------
// TDM: issue a 2D tile load  W[64 rows x CHUNK_K cols, row stride = K floats]
// into LDS at lds_off, with LDS padding of 2 dwords every 64 dwords.
// rem_k = remaining valid columns (tensor_dim0): OOB columns read as zero,
// which natively zero-pads the ragged K=479 edge of the first top layer.
// ---------------------------------------------------------------------------
__device__ __forceinline__ void tdm_issue_tile(const float* gsrc,
                                               unsigned lds_off,
                                               int rem_k, int k_full)
{
    const unsigned long long ga = (unsigned long long)(uintptr_t)gsrc;

    u32x4_t g0;
    g0.x = 1u;                                           // count=1 (valid D#)
    g0.y = lds_off;                                      // LDS byte address
    g0.z = (unsigned)(ga & 0xffffffffu);                 // global_addr[31:0]
    g0.w = (unsigned)((ga >> 32) & 0x01ffffffu)          // global_addr[56:32]
         | (2u << 30);                                   // type = 2 ("image")

    const unsigned data_size    = 2u;   // 4-byte elements
    const unsigned pad_enable   = 1u;
    const unsigned pad_interval = 5u;   // pad every 64 dwords
    const unsigned pad_amount   = 1u;   // pad by 2 dwords
    const unsigned td0   = (unsigned)rem_k;        // tensor_dim0 (valid cols)
    const unsigned td1   = ROWS_PER_BLK;           // tensor_dim1 (rows)
    const unsigned tile0 = CHUNK_K;
    const unsigned tile1 = ROWS_PER_BLK;
    const unsigned long long stride0 = (unsigned long long)k_full;

    i32x8_t g1;
    g1[0] = (int)((data_size << 16) | (pad_enable << 20) |
                  (pad_interval << 22) | (pad_amount << 25));
    g1[1] = (int)((td0 & 0xffffu) << 16);                         // td0[15:0] @ bits 63:48
    g1[2] = (int)(((td0 >> 16) & 0xffffu) | ((td1 & 0xffffu) << 16));
    g1[3] = (int)(((td1 >> 16) & 0xffffu) | (tile0 << 16));
    g1[4] = (int)(tile1 | (0u << 16));                            // tile_dim2 = 0
    g1[5] = (int)(stride0 & 0xffffffffu);
    g1[6] = (int)((stride0 >> 32) & 0xffffu);                     // stride1 = 0
    g1[7] = 0;

    i32x4_t z4 = {};
#if defined(__clang_major__) && (__clang_major__ >= 23)
    i32x8_t z8 = {};
    __builtin_amdgcn_tensor_load_to_lds(g0, g1, z4, z4, z8, 0);   // 6-arg form
#else
    __builtin_amdgcn_tensor_load_to_lds(g0, g1, z4, z4, 0);       // 5-arg form
#endif
}

// ---------------------------------------------------------------------------
// Kernel 1: bottom MLP 13 -> 512 -> 256 -> 128, ReLU each layer. One block.
// ---------------------------------------------------------------------------
__global__ __launch_bounds__(512) void bottom_mlp_kernel(
    const float* __restrict__ x,
    const float* __restrict__ W1, const float* __restrict__ b1,
    const float* __restrict__ W2, const float* __restrict__ b2,
    const float* __restrict__ W3, const float* __restrict__ b3,
    float* __restrict__ xbot)
{
    __shared__ float s0[16];
    __shared__ float s1[512];
    __shared__ float s2[256];
    const int tid = threadIdx.x;

    if (tid < 16) s0[tid] = (tid < 13) ? x[tid] : 0.0f;
    __syncthreads();

    {   // layer 1: 512 outputs, K=13
        float acc = b1[tid];
        #pragma unroll
        for (int k = 0; k < 13; ++k) acc += W1[tid * 13 + k] * s0[k];
        s1[tid] = fmaxf(acc, 0.0f);
    }
    __syncthreads();

    if (tid < 256) {   // layer 2: 256 outputs, K=512
        float acc = b2[tid];
        for (int k = 0; k < 512; ++k) acc += W2[tid * 512 + k] * s1[k];
        s2[tid] = fmaxf(acc, 0.0f);
    }
    __syncthreads();

    if (tid < 128) {   // layer 3: 128 outputs, K=256
        float acc = b3[tid];
        for (int k = 0; k < 256; ++k) acc += W3[tid * 256 + k] * s2[k];
        xbot[tid] = fmaxf(acc, 0.0f);
    }
}

// ---------------------------------------------------------------------------
// Kernel 2: EmbeddingBag sum-pool. Block = table (26), thread = dim (128).
// 512B rows are coalesced across the 128 threads; prefetch next random row.
// ---------------------------------------------------------------------------
__global__ __launch_bounds__(128) void emb_pool_kernel(
    const int* __restrict__ lS_i,
    const float* __restrict__ emb,
    float* __restrict__ pooled)
{
    const int t = blockIdx.x;
    const int d = threadIdx.x;
    const int*   idxs  = lS_i + t * POOL_L;
    const float* table = emb + (size_t)t * NROWS * EMB_D;

    float acc = 0.0f;
    for (int i = 0; i < POOL_L; ++i) {
        const int row = idxs[i];
        if (i + 1 < POOL_L) {
            // hide HBM latency of the random gather (global_prefetch_b8)
            __builtin_prefetch(table + (size_t)idxs[i + 1] * EMB_D + d, 0, 0);
        }
        acc += table[(size_t)row * EMB_D + d];
    }
    pooled[t * EMB_D + d] = acc;
}

// ---------------------------------------------------------------------------
// Kernel 3: dot interaction.  T = [xbot; pooled] padded to 32x128 in LDS,
// Z = T*T^T via 4 tiles of V_WMMA_F32_16X16X4_F32 (K stepped by 4).
// Then R = [xbot(128) | tril(Z,-1) row-major (351) | zero pad to 512].
// Single wave so EXEC is all-ones at every WMMA.
// ---------------------------------------------------------------------------
__global__ __launch_bounds__(32) void interact_wmma_kernel(
    const float* __restrict__ xbot,
    const float* __restrict__ pooled,
    float* __restrict__ R)
{
    __shared__ float T[32][128];
    __shared__ float Z[32][32];
    const int lane = threadIdx.x;      // 0..31
    const int half = lane >> 4;
    const int l16  = lane & 15;

    for (int e = lane; e < 32 * 128; e += 32) {
        const int r = e >> 7, c = e & 127;
        float v = 0.0f;
        if (r == 0)       v = xbot[c];
        else if (r <= 26) v = pooled[(r - 1) * 128 + c];
        T[r][c] = v;
    }
    __syncthreads();

    for (int mi = 0; mi < 2; ++mi) {
        for (int ni = 0; ni < 2; ++ni) {
            v8f_t c = {};
            for (int k0 = 0; k0 < 128; k0 += 4) {
                v2f_t a, b;
                // A 16x4 f32: lanes 0-15 carry K=k0,k0+1; lanes 16-31 K=k0+2,k0+3
                a.x = T[mi * 16 + l16][k0 + 2 * half + 0];
                a.y = T[mi * 16 + l16][k0 + 2 * half + 1];
                // B 4x16 (KxN), B[k][n] = T[ni*16+n][k]
                b.x = T[ni * 16 + l16][k0 + half];
                b.y = T[ni * 16 + l16][k0 + 2 + half];
                c = __builtin_amdgcn_wmma_f32_16x16x4_f32(
                        false, a, false, b, (short)0, c, false, false);
            }
            #pragma unroll
            for (int r = 0; r < 8; ++r)
                Z[mi * 16 + half * 8 + r][ni * 16 + l16] = c[r];
        }
    }
    __syncthreads();

    for (int d = lane; d < 128; d += 32) R[d] = T[0][d];
    // tril_indices(27, k=-1) row-major: pair idx = i*(i-1)/2 + j  (i>j)
    for (int idx = lane; idx < 351; idx += 32) {
        int i = (int)((1.0f + sqrtf(1.0f + 8.0f * (float)idx)) * 0.5f);
        while (i * (i - 1) / 2 > idx)  --i;
        while ((i + 1) * i / 2 <= idx) ++i;
        const int j = idx - i * (i - 1) / 2;
        R[128 + idx] = Z[i][j];
    }
    for (int idx = 479 + lane; idx < 512; idx += 32) R[idx] = 0.0f;  // K-pad
}

// ---------------------------------------------------------------------------
// Kernel 4: top-MLP layer, TDM + WMMA.
// Block = 128 threads (4 waves) = 64 output rows. Wave 0 drives a double-
// buffered TDM pipeline: tensor_load_to_lds pulls a 64x64 f32 weight tile
// (row stride K) into LDS with +2-dword row padding (bank-conflict-free ds
// reads), overlapped with the 4 waves' WMMA on the previous tile. Only batch
// row M=0 of each 16x16 C tile is live; weight traffic is identical to an
// ideal GEMV stream, so the HBM roofline is preserved.
// ---------------------------------------------------------------------------
__global__ __launch_bounds__(128) void top_layer_tdm_wmma_kernel(
    const float* __restrict__ in,    // [>= nchunks*64], zero-padded past K
    const float* __restrict__ W,     // [N, K] row-major
    const float* __restrict__ bias,  // [N]
    float* __restrict__ out,         // [N]
    int K)
{
    __shared__ float Wt[2][ROWS_PER_BLK * LDS_STRIDE];

    const int tid  = threadIdx.x;
    const int wave = tid >> 5;
    const int lane = tid & 31;
    const int half = lane >> 4;
    const int l16  = lane & 15;
    const int r0   = blockIdx.x * ROWS_PER_BLK;
    const int nchunks = (K + CHUNK_K - 1) / CHUNK_K;
    const int myrow = wave * 16 + l16;           // row within the 64-row tile

    const float* wbase = W + (size_t)r0 * K;
    const unsigned lds0 = (unsigned)(uintptr_t)&Wt[0][0];
    const unsigned lds1 = (unsigned)(uintptr_t)&Wt[1][0];

    if (wave == 0) {
        tdm_issue_tile(wbase, lds0, K, K);
        __builtin_amdgcn_s_wait_tensorcnt(0);
    }
    __syncthreads();

    v8f_t c = {};
    for (int ch = 0; ch < nchunks; ++ch) {
        const int buf = ch & 1;
        if (wave == 0 && (ch + 1) < nchunks) {
            tdm_issue_tile(wbase + (ch + 1) * CHUNK_K,
                           (buf ? lds0 : lds1),
                           K - (ch + 1) * CHUNK_K, K);
        }
        const float* wt    = &Wt[buf][myrow * LDS_STRIDE];
        const int    kbase = ch * CHUNK_K;
        #pragma unroll
        for (int k0 = 0; k0 < CHUNK_K; k0 += 4) {
            v2f_t a, b;
            float a0 = 0.0f, a1 = 0.0f;
            if (l16 == 0) {          // M=0 batch row only
                a0 = in[kbase + k0 + 2 * half + 0];
                a1 = in[kbase + k0 + 2 * half + 1];
            }
            a.x = a0; a.y = a1;
            b.x = wt[k0 + half];
            b.y = wt[k0 + 2 + half];
            c = __builtin_amdgcn_wmma_f32_16x16x4_f32(
                    false, a, false, b, (short)0, c, false, false);
        }
        if (wave == 0 && (ch + 1) < nchunks)
            __builtin_amdgcn_s_wait_tensorcnt(0);
        __syncthreads();
    }

    if (half == 0) {     // D row M=0 is VGPR0 of lanes 0..15 (N = lane)
        const int orow = r0 + myrow;
        out[orow] = fmaxf(c[0] + bias[orow], 0.0f);
    }
}

// ---------------------------------------------------------------------------
// Kernel 5: final 512 -> 1 + sigmoid.
// ---------------------------------------------------------------------------
__global__ __launch_bounds__(128) void top_final_kernel(
    const float* __restrict__ in,   // [512]
    const float* __restrict__ W,    // [1,512]
    const float* __restrict__ b,    // [1]
    float* __restrict__ out)        // [1]
{
    __shared__ float red[4];
    const int tid = threadIdx.x;
    float acc = 0.0f;
    for (int k = tid; k < 512; k += 128) acc += in[k] * W[k];
    for (int off = 16; off > 0; off >>= 1)
        acc += __shfl_down(acc, off, 32);
    if ((tid & 31) == 0) red[tid >> 5] = acc;
    __syncthreads();
    if (tid == 0) {
        const float s = red[0] + red[1] + red[2] + red[3] + b[0];
        out[0] = 1.0f / (1.0f + __expf(-s));
    }
}

// ---------------------------------------------------------------------------
// Launch
// ---------------------------------------------------------------------------
extern "C" void kernel_launch(void* const* d_in, const int* in_sizes, int n_in,
                              void* d_out, int out_size, void* d_ws, size_t ws_size,
                              hipStream_t stream)
{
    (void)in_sizes; (void)n_in; (void)out_size; (void)ws_size;

    const float* dense_x = (const float*)d_in[0];
    const int*   lS_i    = (const int*)  d_in[1];
    const float* emb_w   = (const float*)d_in[2];
    const float* bW1 = (const float*)d_in[3];  const float* bb1 = (const float*)d_in[4];
    const float* bW2 = (const float*)d_in[5];  const float* bb2 = (const float*)d_in[6];
    const float* bW3 = (const float*)d_in[7];  const float* bb3 = (const float*)d_in[8];
    const float* tW1 = (const float*)d_in[9];  const float* tb1 = (const float*)d_in[10];
    const float* tW2 = (const float*)d_in[11]; const float* tb2 = (const float*)d_in[12];
    const float* tW3 = (const float*)d_in[13]; const float* tb3 = (const float*)d_in[14];
    const float* tW4 = (const float*)d_in[15]; const float* tb4 = (const float*)d_in[16];
    float* out = (float*)d_out;

    // workspace layout (floats)
    float* ws     = (float*)d_ws;
    float* xbot   = ws;              // [128]
    float* pooled = ws + 128;        // [26*128]
    float* R      = ws + 3456;       // [512]  (479 used + zero pad)
    float* h1     = ws + 3968;       // [1024]
    float* h2     = ws + 4992;       // [1024]
    float* h3     = ws + 6016;       // [512]

    bottom_mlp_kernel<<<1, 512, 0, stream>>>(dense_x, bW1, bb1, bW2, bb2, bW3, bb3, xbot);
    emb_pool_kernel<<<NTAB, EMB_D, 0, stream>>>(lS_i, emb_w, pooled);
    interact_wmma_kernel<<<1, 32, 0, stream>>>(xbot, pooled, R);

    top_layer_tdm_wmma_kernel<<<1024 / ROWS_PER_BLK, 128, 0, stream>>>(R,  tW1, tb1, h1, 479);
    top_layer_tdm_wmma_kernel<<<1024 / ROWS_PER_BLK, 128, 0, stream>>>(h1, tW2, tb2, h2, 1024);
    top_layer_tdm_wmma_kernel<<< 512 / ROWS_PER_BLK, 128, 0, stream>>>(h2, tW3, tb3, h3, 1024);
    top_final_kernel<<<1, 128, 0, stream>>>(h3, tW4, tb4, out);
}